// RegionProposalNetwork_75230647156940
// MI455X (gfx1250) — compile-verified
//
#include <hip/hip_runtime.h>
#include <math.h>
#include <float.h>

// ---------------- problem constants (match reference) ----------------
#define B_IMGS      16
#define NUM_LEVELS  5
#define A_TOTAL     261888
#define K_TOT       4768          // 1000+1000+1000+1000+768
#define POST_N      1000
#define NMS_THR     0.7f
#define MIN_SIZE_F  0.001f
#define SCORE_THR   0.0f
#define BBOX_CLIP   4.135166556742356f   // log(1000/16)
#define NEGV        (-1e9f)

#define SORT_N      8192          // next pow2 >= K_TOT for bitonic sort
#define NMS_T       1024
#define SEL_T       1024
#define TILE        4096          // floats staged per LDS tile (16KB)
#define TIE_MAX     1024
#define KEEP_WORDS  ((K_TOT + 31) / 32)

__device__ const int d_LVL_N[NUM_LEVELS]    = {196608, 49152, 12288, 3072, 768};
__device__ const int d_LVL_OFF[NUM_LEVELS]  = {0, 196608, 245760, 258048, 261120};
__device__ const int d_LVL_K[NUM_LEVELS]    = {1000, 1000, 1000, 1000, 768};
__device__ const int d_SLOT_OFF[NUM_LEVELS] = {0, 1000, 2000, 3000, 4000};

// ---------------- CDNA5 feature probes ----------------
#if defined(__has_builtin)
#  if __has_builtin(__builtin_amdgcn_global_load_async_to_lds_b128)
#    define USE_ASYNC_LDS 1
#  else
#    define USE_ASYNC_LDS 0
#  endif
#  if __has_builtin(__builtin_amdgcn_s_wait_asynccnt)
#    define HAVE_WAIT_ASYNC 1
#  else
#    define HAVE_WAIT_ASYNC 0
#  endif
#  if __has_builtin(__builtin_amdgcn_tensor_load_to_lds) && \
      __has_builtin(__builtin_amdgcn_s_wait_tensorcnt)
#    define USE_TDM 1
#  else
#    define USE_TDM 0
#  endif
#else
#  define USE_ASYNC_LDS 0
#  define HAVE_WAIT_ASYNC 0
#  define USE_TDM 0
#endif

typedef int v4i_t __attribute__((ext_vector_type(4)));
#if USE_ASYNC_LDS
typedef __attribute__((address_space(1))) v4i_t* g_v4i_p;
typedef __attribute__((address_space(3))) v4i_t* l_v4i_p;
#endif
#if USE_TDM
typedef unsigned u32x4_t __attribute__((ext_vector_type(4)));
typedef int i32x8_t __attribute__((ext_vector_type(8)));
typedef int i32x4_t __attribute__((ext_vector_type(4)));
#endif

// Map float -> order-preserving unsigned (IEEE total order for our purposes)
__device__ __forceinline__ unsigned ordf(float f) {
    unsigned x = __float_as_uint(f);
    return (x & 0x80000000u) ? ~x : (x | 0x80000000u);
}

__device__ __forceinline__ int slot_level(int s) {
    return (s < 1000) ? 0 : (s < 2000) ? 1 : (s < 3000) ? 2 : (s < 4000) ? 3 : 4;
}

// Per-lane async global->LDS staging (ASYNCcnt-tracked).
__device__ __forceinline__ void stage_async(const float* g, float* lds, int cnt) {
#if USE_ASYNC_LDS
    {
        float* gnc = const_cast<float*>(g);   // builtin param is non-const
        for (int i = (int)threadIdx.x * 4; i < cnt; i += (int)blockDim.x * 4) {
            __builtin_amdgcn_global_load_async_to_lds_b128(
                (g_v4i_p)(gnc + i),
                (l_v4i_p)(lds + i),
                0 /*offset*/, 0 /*cpol*/);
        }
    }
#  if HAVE_WAIT_ASYNC
    __builtin_amdgcn_s_wait_asynccnt(0);
#  else
    asm volatile("s_wait_asynccnt 0" ::: "memory");
#  endif
#else
    for (int i = (int)threadIdx.x; i < cnt; i += (int)blockDim.x) lds[i] = g[i];
#endif
}

// Stage `cnt` floats (cnt % 4 == 0, 16B aligned) from global into LDS.
// Full tiles go through the Tensor Data Mover (one descriptor, TENSORcnt);
// partial tiles use per-lane async loads (ASYNCcnt).
__device__ __forceinline__ void stage_tile(const float* g, float* lds, int cnt) {
    __syncthreads();   // previous consumers of `lds` are done
#if USE_TDM
    if (cnt == TILE) {   // uniform per block
        if (threadIdx.x == 0) {
            unsigned lds_off =
                (unsigned)(size_t)(__attribute__((address_space(3))) char*)lds;
            unsigned long long ga = (unsigned long long)(const void*)g;
            // D# group 0: count=1 | lds_addr | global_addr[56:0] | type=2
            u32x4_t g0;
            g0.x = 1u;                                   // 1 valid descriptor
            g0.y = lds_off;                              // bits 63:32
            g0.z = (unsigned)(ga & 0xFFFFFFFFull);       // bits 95:64
            g0.w = (unsigned)((ga >> 32) & 0x01FFFFFFull) | (2u << 30); // type=2
            // D# group 1: 1 row of TILE 4-byte elements
            i32x8_t g1;
            g1[0] = (2 << 16);                 // workgroup_mask=0, data_size=4B
            g1[1] = (TILE & 0xFFFF) << 16;     // tensor_dim0[15:0]
            g1[2] = ((TILE >> 16) & 0xFFFF)    // tensor_dim0[31:16]
                    | (1 << 16);               // tensor_dim1 = 1
            g1[3] = (TILE & 0xFFFF) << 16;     // tile_dim0 = TILE
            g1[4] = 1;                         // tile_dim1 = 1, tile_dim2 = 0
            g1[5] = TILE;                      // tensor_dim0_stride[31:0]
            g1[6] = 0;
            g1[7] = 0;
            i32x4_t gz4 = {0, 0, 0, 0};        // groups 2/3 unused (<=2D)
            i32x8_t gz8 = {0, 0, 0, 0, 0, 0, 0, 0};
            // 6-arg form on this toolchain (probe-confirmed arity)
            __builtin_amdgcn_tensor_load_to_lds(g0, g1, gz4, gz4, gz8,
                                                0 /*cpol*/);
        }
        __builtin_amdgcn_s_wait_tensorcnt(0);
    } else
#endif
    {
        stage_async(g, lds, cnt);
    }
    __syncthreads();   // tile visible to whole block
}

// =====================================================================
// Kernel 1: exact per-(image, level) top-k via 4-pass radix select.
// One block per (b, level). Writes selected anchor indices into ws_sel.
// =====================================================================
__global__ __launch_bounds__(SEL_T)
void select_topk_kernel(const float* __restrict__ obj, int* __restrict__ ws_sel) {
    __shared__ __align__(16) float tile[TILE];
    __shared__ unsigned hist[256];
    __shared__ int ties[TIE_MAX];
    __shared__ unsigned sh_prefix;
    __shared__ int sh_kneed, sh_scnt, sh_tcnt;

    const int blk = blockIdx.x;
    const int b = blk / NUM_LEVELS, l = blk % NUM_LEVELS;
    const int n = d_LVL_N[l];
    const int off = d_LVL_OFF[l];
    const int k = d_LVL_K[l];
    const int slotbase = b * K_TOT + d_SLOT_OFF[l];
    const float* src = obj + (size_t)b * A_TOTAL + off;
    const int tid = threadIdx.x;

    if (tid == 0) { sh_prefix = 0u; sh_kneed = k; sh_scnt = 0; sh_tcnt = 0; }
    // safe defaults so every slot holds a valid anchor index no matter what
    for (int i = tid; i < k; i += SEL_T) ws_sel[slotbase + i] = off;
    __syncthreads();

    // ---- 4 radix passes: find k-th largest ordered-uint value ----
    for (int pass = 0; pass < 4; ++pass) {
        const int shift = 24 - 8 * pass;
        for (int i = tid; i < 256; i += SEL_T) hist[i] = 0u;
        __syncthreads();
        const unsigned prefix = sh_prefix;
        const unsigned mask_hi = pass ? (0xFFFFFFFFu << (32 - 8 * pass)) : 0u;

        for (int t0 = 0; t0 < n; t0 += TILE) {
            const int cnt = (n - t0 < TILE) ? (n - t0) : TILE;
            stage_tile(src + t0, tile, cnt);
            for (int i = tid; i < cnt; i += SEL_T) {
                unsigned u = ordf(tile[i]);
                if ((u & mask_hi) == (prefix & mask_hi))
                    atomicAdd(&hist[(u >> shift) & 255u], 1u);
            }
        }
        __syncthreads();   // histogram complete

        if (tid == 0) {
            int kneed = sh_kneed;
            unsigned cum = 0;
            for (int q = 255; q >= 0; --q) {
                unsigned c = hist[q];
                if (cum + c >= (unsigned)kneed) {
                    sh_prefix = prefix | ((unsigned)q << shift);
                    sh_kneed = kneed - (int)cum;   // >= 1
                    break;
                }
                cum += c;
            }
        }
        __syncthreads();
    }

    const unsigned thr = sh_prefix;   // exact k-th largest value (ordered bits)
    // ---- selection pass: all > thr, plus (k - count(>thr)) of == thr ----
    for (int t0 = 0; t0 < n; t0 += TILE) {
        const int cnt = (n - t0 < TILE) ? (n - t0) : TILE;
        stage_tile(src + t0, tile, cnt);
        for (int i = tid; i < cnt; i += SEL_T) {
            unsigned u = ordf(tile[i]);
            if (u > thr) {
                int p = atomicAdd(&sh_scnt, 1);
                if (p < k) ws_sel[slotbase + p] = off + t0 + i;
            } else if (u == thr) {
                int p = atomicAdd(&sh_tcnt, 1);
                if (p < TIE_MAX) ties[p] = off + t0 + i;
            }
        }
    }
    __syncthreads();

    if (tid == 0) {   // ties by lowest index first (matches lax.top_k)
        int m = sh_tcnt < TIE_MAX ? sh_tcnt : TIE_MAX;
        for (int a = 1; a < m; ++a) {
            int v = ties[a];
            int p = a - 1;
            while (p >= 0 && ties[p] > v) { ties[p + 1] = ties[p]; --p; }
            ties[p + 1] = v;
        }
    }
    __syncthreads();

    int c = sh_scnt < k ? sh_scnt : k;
    int need = k - c;
    int avail = sh_tcnt < TIE_MAX ? sh_tcnt : TIE_MAX;
    for (int i = tid; i < need; i += SEL_T)
        if (i < avail) ws_sel[slotbase + c + i] = ties[i];
}

// =====================================================================
// Kernel 2: decode + clip + sigmoid + validity for selected anchors.
// =====================================================================
__global__ void decode_kernel(const float* __restrict__ obj,
                              const float* __restrict__ deltas,
                              const float* __restrict__ anchors,
                              const int* __restrict__ ws_sel,
                              float* __restrict__ ws_boxes,
                              float* __restrict__ ws_score,
                              int* __restrict__ ws_valid,
                              const int* __restrict__ p_ih,
                              const int* __restrict__ p_iw) {
    int gid = blockIdx.x * blockDim.x + threadIdx.x;
    if (gid >= B_IMGS * K_TOT) return;
    int b = gid / K_TOT;
    int a = ws_sel[gid];

    const float* dp = deltas + ((size_t)b * A_TOTAL + (size_t)a) * 4;
    const float* ap = anchors + (size_t)a * 4;
    __builtin_prefetch(dp, 0, 0);   // gfx1250: global_prefetch_b8
    __builtin_prefetch(ap, 0, 0);

    const float fw = (float)*p_iw, fh = (float)*p_ih;

    float ax1 = ap[0], ay1 = ap[1], ax2 = ap[2], ay2 = ap[3];
    float w = ax2 - ax1, h = ay2 - ay1;
    float cx = ax1 + 0.5f * w, cy = ay1 + 0.5f * h;

    float dx = dp[0], dy = dp[1];
    float dw = fminf(dp[2], BBOX_CLIP), dh = fminf(dp[3], BBOX_CLIP);
    float pcx = dx * w + cx, pcy = dy * h + cy;
    float pw = expf(dw) * w, ph = expf(dh) * h;

    float x1 = pcx - 0.5f * pw, y1 = pcy - 0.5f * ph;
    float x2 = pcx + 0.5f * pw, y2 = pcy + 0.5f * ph;
    x1 = fminf(fmaxf(x1, 0.f), fw);
    x2 = fminf(fmaxf(x2, 0.f), fw);
    y1 = fminf(fmaxf(y1, 0.f), fh);
    y2 = fminf(fmaxf(y2, 0.f), fh);

    float o = obj[(size_t)b * A_TOTAL + (size_t)a];
    float sc = 1.f / (1.f + expf(-o));
    int valid = ((x2 - x1) >= MIN_SIZE_F) && ((y2 - y1) >= MIN_SIZE_F) &&
                (sc >= SCORE_THR);

    float* bo = ws_boxes + (size_t)gid * 4;
    bo[0] = x1; bo[1] = y1; bo[2] = x2; bo[3] = y2;
    ws_score[gid] = sc;
    ws_valid[gid] = valid;
}

// =====================================================================
// Kernel 3: per-image sort + greedy NMS + compaction into d_out.
// One block (1024 threads) per image; ~143KB LDS (< 320KB/WGP).
// =====================================================================
__global__ __launch_bounds__(NMS_T)
void nms_out_kernel(const float* __restrict__ ws_boxes,
                    const float* __restrict__ ws_score,
                    const int* __restrict__ ws_valid,
                    float* __restrict__ out,
                    const int* __restrict__ p_ih,
                    const int* __restrict__ p_iw) {
    __shared__ float skey[SORT_N];
    __shared__ int sidx[SORT_N];
    __shared__ float4 bb[K_TOT];
    __shared__ unsigned keepw[KEEP_WORDS];
    __shared__ int scanbuf[NMS_T];

    const int b = blockIdx.x;
    const int tid = threadIdx.x;
    const float sep = fmaxf((float)*p_iw, (float)*p_ih) + 1.0f;

    float* out_boxes = out;                                   // [B][1000][4]
    float* out_scores = out + (size_t)B_IMGS * POST_N * 4;    // [B][1000]

    // zero-fill this image's outputs (harness poisons d_out)
    for (int i = tid; i < POST_N; i += NMS_T) {
        out_scores[b * POST_N + i] = 0.f;
        float* ob = out_boxes + ((size_t)b * POST_N + i) * 4;
        ob[0] = 0.f; ob[1] = 0.f; ob[2] = 0.f; ob[3] = 0.f;
    }
    __threadfence();
    __syncthreads();

    const float* mybox = ws_boxes + (size_t)b * K_TOT * 4;
    const float* mysc = ws_score + (size_t)b * K_TOT;
    const int* myval = ws_valid + (size_t)b * K_TOT;

    // load (masked score, slot) pairs; pad with -FLT_MAX
    for (int i = tid; i < SORT_N; i += NMS_T) {
        skey[i] = (i < K_TOT) ? (myval[i] ? mysc[i] : NEGV) : -FLT_MAX;
        sidx[i] = i;
    }
    __syncthreads();

    // bitonic sort, descending by (key, idx ascending) -> deterministic
    for (unsigned kk = 2; kk <= SORT_N; kk <<= 1) {
        for (unsigned j = kk >> 1; j > 0; j >>= 1) {
            for (unsigned i = tid; i < SORT_N; i += NMS_T) {
                unsigned ixj = i ^ j;
                if (ixj > i) {
                    float ka = skey[i], kb2 = skey[ixj];
                    int ia = sidx[i], ib = sidx[ixj];
                    bool before = (ka > kb2) || (ka == kb2 && ia < ib);
                    bool doSwap = ((i & kk) == 0) ? (!before) : before;
                    if (doSwap) {
                        skey[i] = kb2; skey[ixj] = ka;
                        sidx[i] = ib;  sidx[ixj] = ia;
                    }
                }
            }
            __syncthreads();
        }
    }

    // gather level-offset boxes in sorted order; init keep = valid
    for (int w = tid; w < KEEP_WORDS; w += NMS_T) keepw[w] = 0u;
    __syncthreads();
    for (int i = tid; i < K_TOT; i += NMS_T) {
        int s = sidx[i];
        const float* bp = mybox + (size_t)s * 4;
        float offv = (float)slot_level(s) * sep;
        bb[i] = make_float4(bp[0] + offv, bp[1] + offv, bp[2] + offv, bp[3] + offv);
        if (skey[i] > -5e8f)   // valid entries (NEGV and padding excluded)
            atomicOr(&keepw[i >> 5], 1u << (i & 31));
    }
    __syncthreads();

    // greedy NMS: serial pivots, parallel suppression (torchvision semantics)
    for (int i = 0; i < K_TOT; ++i) {
        bool alive = (keepw[i >> 5] >> (i & 31)) & 1u;   // uniform across block
        if (alive) {
            float4 bi = bb[i];
            float ai = fmaxf(bi.z - bi.x, 0.f) * fmaxf(bi.w - bi.y, 0.f);
            for (int j = i + 1 + tid; j < K_TOT; j += NMS_T) {
                if ((keepw[j >> 5] >> (j & 31)) & 1u) {
                    float4 bj = bb[j];
                    float xx1 = fmaxf(bi.x, bj.x), yy1 = fmaxf(bi.y, bj.y);
                    float xx2 = fminf(bi.z, bj.z), yy2 = fminf(bi.w, bj.w);
                    float inter = fmaxf(xx2 - xx1, 0.f) * fmaxf(yy2 - yy1, 0.f);
                    float aj = fmaxf(bj.z - bj.x, 0.f) * fmaxf(bj.w - bj.y, 0.f);
                    float iou = inter / (ai + aj - inter + 1e-9f);
                    if (iou > NMS_THR)
                        atomicAnd(&keepw[j >> 5], ~(1u << (j & 31)));
                }
            }
        }
        __syncthreads();
    }

    // compaction: kept entries are already in descending score order
    const int CHUNK = (K_TOT + NMS_T - 1) / NMS_T;   // 5
    int base = tid * CHUNK;
    int cnt = 0;
    for (int r = 0; r < CHUNK; ++r) {
        int i = base + r;
        if (i < K_TOT && ((keepw[i >> 5] >> (i & 31)) & 1u)) ++cnt;
    }
    scanbuf[tid] = cnt;
    __syncthreads();
    for (int d = 1; d < NMS_T; d <<= 1) {
        int v = (tid >= d) ? scanbuf[tid - d] : 0;
        __syncthreads();
        scanbuf[tid] += v;
        __syncthreads();
    }
    int rank = scanbuf[tid] - cnt;   // exclusive prefix for this chunk
    for (int r = 0; r < CHUNK; ++r) {
        int i = base + r;
        if (i < K_TOT && ((keepw[i >> 5] >> (i & 31)) & 1u)) {
            if (rank < POST_N) {
                int s = sidx[i];
                const float* bp = mybox + (size_t)s * 4;
                float* ob = out_boxes + ((size_t)b * POST_N + rank) * 4;
                ob[0] = bp[0]; ob[1] = bp[1]; ob[2] = bp[2]; ob[3] = bp[3];
                out_scores[b * POST_N + rank] = skey[i];
            }
            ++rank;
        }
    }
}

// =====================================================================
// Launch
// =====================================================================
extern "C" void kernel_launch(void* const* d_in, const int* in_sizes, int n_in,
                              void* d_out, int out_size, void* d_ws, size_t ws_size,
                              hipStream_t stream) {
    (void)in_sizes; (void)n_in; (void)out_size; (void)ws_size;
    const float* objectness = (const float*)d_in[0];
    const float* deltas     = (const float*)d_in[1];
    const float* anchors    = (const float*)d_in[2];
    const int*   p_ih       = (const int*)d_in[3];
    const int*   p_iw       = (const int*)d_in[4];
    float* out = (float*)d_out;

    // workspace layout (bytes): sel | boxes | score | valid  (~2.1 MB total)
    char* ws = (char*)d_ws;
    int*   ws_sel   = (int*)ws;
    float* ws_boxes = (float*)(ws + (size_t)B_IMGS * K_TOT * sizeof(int));
    float* ws_score = ws_boxes + (size_t)B_IMGS * K_TOT * 4;
    int*   ws_valid = (int*)(ws_score + (size_t)B_IMGS * K_TOT);

    select_topk_kernel<<<dim3(B_IMGS * NUM_LEVELS), dim3(SEL_T), 0, stream>>>(
        objectness, ws_sel);

    int total = B_IMGS * K_TOT;
    decode_kernel<<<dim3((total + 255) / 256), dim3(256), 0, stream>>>(
        objectness, deltas, anchors, ws_sel, ws_boxes, ws_score, ws_valid,
        p_ih, p_iw);

    nms_out_kernel<<<dim3(B_IMGS), dim3(NMS_T), 0, stream>>>(
        ws_boxes, ws_score, ws_valid, out, p_ih, p_iw);
}